// RboxSingleRoIExtractor_2327872274504
// MI455X (gfx1250) — compile-verified
//
#include <hip/hip_runtime.h>
#include <hip/hip_bf16.h>

#define OUTSZ 7
#define SGRID 14      // OUTSZ * 2 samples
#define NSAMP 196     // 14*14
#define NBIN 49
#define CG 64         // channels per block
#define CALL 256      // total channels
#define VPAD 65       // padded sample stride in LDS (floats) -> conflict-free banks

typedef float v2f __attribute__((ext_vector_type(2)));
typedef float v8f __attribute__((ext_vector_type(8)));

struct SampleMeta {
    int o00, o01, o10, o11;
    float w00, w01, w10, w11;
};

__global__ __launch_bounds__(256) void rroi_extract_wmma_kernel(
    const float* __restrict__ f0, const float* __restrict__ f1,
    const float* __restrict__ f2, const float* __restrict__ f3,
    const float* __restrict__ rois, float* __restrict__ out)
{
    __shared__ float Vs[NSAMP * VPAD];       // bin-major samples: [slot][ch], padded
    __shared__ SampleMeta meta[NSAMP];       // per-sample gather metadata (slot order)

    const int tid   = threadIdx.x;
    const int roi   = blockIdx.x >> 2;       // 4 channel groups per roi
    const int cg    = blockIdx.x & 3;
    const int cbase = cg * CG;

    // ---- uniform per-roi parameters (every thread computes; wave-uniform) ----
    const float rb  = rois[roi * 6 + 0];
    const float rcx = rois[roi * 6 + 1];
    const float rcy = rois[roi * 6 + 2];
    const float rw  = rois[roi * 6 + 3];
    const float rh  = rois[roi * 6 + 4];
    const float rth = rois[roi * 6 + 5];

    const float scl = sqrtf(rw * rh);
    int lvl = (int)floorf(log2f(scl * (1.0f / 56.0f) + 1e-6f));
    lvl = lvl < 0 ? 0 : (lvl > 3 ? 3 : lvl);
    const int   Wd = 256 >> lvl;             // square maps: H == W
    const int   HW = Wd * Wd;
    const float sp = 1.0f / (float)(4 << lvl);
    const float* lbase = (lvl == 0) ? f0 : (lvl == 1) ? f1 : (lvl == 2) ? f2 : f3;
    const float* base  = lbase + (size_t)((int)rb) * CALL * HW + (size_t)cbase * HW;

    // ---- Stage A: per-sample coordinates, bilinear weights, bin-major slots ----
    if (tid < NSAMP) {
        const int sy = tid / SGRID;
        const int sx = tid - sy * SGRID;
        const float cx = rcx * sp, cy = rcy * sp;
        const float w  = rw * sp,  h  = rh * sp;
        const float ct = cosf(rth), st = sinf(rth);
        const float fx = ((sx + 0.5f) * (1.0f / SGRID) - 0.5f) * w;
        const float fy = ((sy + 0.5f) * (1.0f / SGRID) - 0.5f) * h;
        const float X = cx + fx * ct - fy * st;
        const float Y = cy + fx * st + fy * ct;
        const float Wf = (float)Wd;
        const bool valid = (X > -1.0f) && (X < Wf) && (Y > -1.0f) && (Y < Wf);
        const float x = fminf(fmaxf(X, 0.0f), Wf - 1.0f);
        const float y = fminf(fmaxf(Y, 0.0f), Wf - 1.0f);
        const int x0 = (int)floorf(x);
        const int y0 = (int)floorf(y);
        const int x1 = min(x0 + 1, Wd - 1);
        const int y1 = min(y0 + 1, Wd - 1);
        const float lx = x - (float)x0, ly = y - (float)y0;
        const float hx = 1.0f - lx,     hy = 1.0f - ly;
        const float vf = valid ? 1.0f : 0.0f;

        const int bin = (sy >> 1) * OUTSZ + (sx >> 1);
        const int j   = ((sy & 1) << 1) | (sx & 1);
        SampleMeta m;
        m.o00 = y0 * Wd + x0;  m.o01 = y0 * Wd + x1;
        m.o10 = y1 * Wd + x0;  m.o11 = y1 * Wd + x1;
        m.w00 = hy * hx * vf;  m.w01 = hy * lx * vf;
        m.w10 = ly * hx * vf;  m.w11 = ly * lx * vf;
        meta[bin * 4 + j] = m;               // store in slot (bin-major) order
    }
    __syncthreads();

    // ---- Stage B: gather bilinear samples into LDS ----
    // Lanes walk consecutive samples of the SAME channel: adjacent samples are
    // spatially adjacent -> lanes share cachelines (fewer L2 line requests).
    for (int e = tid; e < NSAMP * CG; e += 256) {
        const int slot = e % NSAMP;          // 0..195 (bin*4 + j), consecutive per lane
        const int c    = e / NSAMP;          // channel within group
        const SampleMeta m = meta[slot];
        const float* p = base + (size_t)c * HW;
        const float v = p[m.o00] * m.w00 + p[m.o01] * m.w01 +
                        p[m.o10] * m.w10 + p[m.o11] * m.w11;
        Vs[slot * VPAD + c] = v;             // bank = (slot + c) & 63: conflict-free
    }
    __syncthreads();

    // ---- Stage C: WMMA pooling  O[16ch,16bin] = V[16ch,K] x P[K,16bin] ----
    // V_WMMA_F32_16X16X4_F32; each K=4 step consumes one bin's 4 samples with
    // B[k,n] = 0.25 * (n == t) (identical across k-rows -> B layout-proof).
    // Scalarize tile indices so control flow is uniform (EXEC all-1s at WMMA).
    const int waveU = __builtin_amdgcn_readfirstlane(tid >> 5);
    const int lane  = tid & 31;
    const int mrow  = lane & 15;             // A-matrix M index / D column N index
    const int half  = lane >> 4;             // 0: K=0/1   1: K=2/3 halves of A

    #pragma unroll
    for (int pp = 0; pp < 2; ++pp) {
        const int p  = waveU + pp * 8;       // SGPR tile index
        const int mt = p & 3;
        const int nt = p >> 2;
        const int chA = mt * 16 + mrow;      // channel row this lane feeds into A
        v8f acc = {0.f, 0.f, 0.f, 0.f, 0.f, 0.f, 0.f, 0.f};
        const int tmax = (nt < 3) ? 16 : 1;  // bins 48..63: only bin 48 exists
        for (int t = 0; t < tmax; ++t) {     // uniform trip count (scalar)
            const int slot = (nt * 16 + t) * 4 + 2 * half;
            v2f a;
            a.x = Vs[slot * VPAD + chA];         // K = 2*half
            a.y = Vs[(slot + 1) * VPAD + chA];   // K = 2*half + 1
            const float bval = (mrow == t) ? 0.25f : 0.0f;
            v2f b; b.x = bval; b.y = bval;
            acc = __builtin_amdgcn_wmma_f32_16x16x4_f32(
                false, a, false, b, (short)0, acc, false, false);
        }
        // D layout: lane gives N = lane&15; VGPR r gives M = r + 8*half
        const int bin = nt * 16 + mrow;
        if (bin < NBIN) {
            #pragma unroll
            for (int r = 0; r < 8; ++r) {
                const int och = cbase + mt * 16 + r + 8 * half;
                out[((size_t)roi * CALL + och) * NBIN + bin] = acc[r];
            }
        }
    }
}

extern "C" void kernel_launch(void* const* d_in, const int* in_sizes, int n_in,
                              void* d_out, int out_size, void* d_ws, size_t ws_size,
                              hipStream_t stream) {
    const float* f0   = (const float*)d_in[0];
    const float* f1   = (const float*)d_in[1];
    const float* f2   = (const float*)d_in[2];
    const float* f3   = (const float*)d_in[3];
    const float* rois = (const float*)d_in[4];
    float* out = (float*)d_out;

    const int K = in_sizes[4] / 6;           // number of rois
    dim3 grid(K * 4);                        // (roi, 64-channel group)
    dim3 block(256);
    rroi_extract_wmma_kernel<<<grid, block, 0, stream>>>(f0, f1, f2, f3, rois, out);
}